// MessageBlock_9096740733260
// MI455X (gfx1250) — compile-verified
//
#include <hip/hip_runtime.h>

typedef __attribute__((ext_vector_type(2))) float v2f;
typedef __attribute__((ext_vector_type(8))) float v8f;

#define DD 64
#define HH 128
#define SX_STRIDE 68    // 64 + 4 pad: bank-conflict-free fragment reads
#define SH_STRIDE 132   // 128 + 4 pad

// One wave computes Y[row0:row0+16] = relu(X @ W1 + b1) @ W2 + b2 via f32 WMMA.
__global__ __launch_bounds__(32)
void mlp16_wmma_kernel(const float* __restrict__ X,
                       const float* __restrict__ W1,
                       const float* __restrict__ b1,
                       const float* __restrict__ W2,
                       const float* __restrict__ b2,
                       float* __restrict__ Y)
{
    __shared__ float sX[16 * SX_STRIDE];
    __shared__ float sH[16 * SH_STRIDE];

    const int lane = threadIdx.x;       // 0..31 (wave32)
    const int m    = lane & 15;         // row / column within 16-tile
    const int kh   = lane >> 4;         // K half-select (ISA A/B layout)
    const long row0 = (long)blockIdx.x * 16;

    // Stage 16x64 X tile into LDS (float4, coalesced).
    #pragma unroll
    for (int i = 0; i < 8; ++i) {
        int idx = lane + i * 32;        // 256 float4 chunks
        int r   = idx >> 4;
        int c4  = idx & 15;
        float4 v = ((const float4*)(X + (row0 + r) * DD))[c4];
        float* d = &sX[r * SX_STRIDE + c4 * 4];
        d[0] = v.x; d[1] = v.y; d[2] = v.z; d[3] = v.w;
    }
    __syncthreads();

    // GEMM1: H = relu(Xtile[16x64] @ W1[64x128] + b1) -> sH
    #pragma unroll
    for (int nt = 0; nt < HH / 16; ++nt) {
        const int h0 = nt * 16;
        const float bias = b1[h0 + m];
        v8f c;
        #pragma unroll
        for (int g = 0; g < 8; ++g) c[g] = bias;
        #pragma unroll
        for (int k = 0; k < DD / 4; ++k) {
            const int kb = k * 4 + 2 * kh;
            v2f a = *(const v2f*)&sX[m * SX_STRIDE + kb];   // A: 16x4 f32 layout
            v2f b;
            b[0] = W1[(long)kb * HH + h0 + m];              // B: 4x16 f32 layout
            b[1] = W1[(long)(kb + 1) * HH + h0 + m];
            c = __builtin_amdgcn_wmma_f32_16x16x4_f32(false, a, false, b,
                                                      (short)0, c, false, false);
        }
        // D layout: VGPR g -> (M = g + 8*kh, N = m); apply relu, park in LDS
        #pragma unroll
        for (int g = 0; g < 8; ++g) {
            float v = c[g] > 0.0f ? c[g] : 0.0f;
            sH[(g + kh * 8) * SH_STRIDE + h0 + m] = v;
        }
    }
    __syncthreads();

    // GEMM2: Y = H[16x128] @ W2[128x64] + b2
    #pragma unroll
    for (int nt = 0; nt < DD / 16; ++nt) {
        const int j0 = nt * 16;
        const float bias = b2[j0 + m];
        v8f c;
        #pragma unroll
        for (int g = 0; g < 8; ++g) c[g] = bias;
        #pragma unroll
        for (int k = 0; k < HH / 4; ++k) {
            const int kb = k * 4 + 2 * kh;
            v2f a = *(const v2f*)&sH[m * SH_STRIDE + kb];
            v2f b;
            b[0] = W2[(long)kb * DD + j0 + m];
            b[1] = W2[(long)(kb + 1) * DD + j0 + m];
            c = __builtin_amdgcn_wmma_f32_16x16x4_f32(false, a, false, b,
                                                      (short)0, c, false, false);
        }
        #pragma unroll
        for (int g = 0; g < 8; ++g) {
            Y[(row0 + g + kh * 8) * DD + j0 + m] = c[g];
        }
    }
}

__global__ void zero_kernel(float4* __restrict__ p, long n4) {
    long i = (long)blockIdx.x * blockDim.x + threadIdx.x;
    if (i < n4) p[i] = make_float4(0.f, 0.f, 0.f, 0.f);
}

// 16 lanes per edge: float4 gather of Y[src], atomic f32 adds into agg[dst].
__global__ __launch_bounds__(256)
void scatter_kernel(const float* __restrict__ Y,
                    const int* __restrict__ src,
                    const int* __restrict__ dst,
                    float* __restrict__ agg, int nE)
{
    int t = blockIdx.x * 256 + threadIdx.x;
    int e = t >> 4;
    int l = t & 15;
    if (e >= nE) return;
    int s = src[e];
    int d = dst[e];
    float4 v = *(const float4*)(Y + (long)s * DD + l * 4);
    float* p = agg + (long)d * DD + l * 4;
    atomicAdd(p + 0, v.x);
    atomicAdd(p + 1, v.y);
    atomicAdd(p + 2, v.z);
    atomicAdd(p + 3, v.w);
}

// out = relu(x_v + agg), in place on d_out.
__global__ void finalize_kernel(const float4* __restrict__ xv,
                                float4* __restrict__ out, long n4)
{
    long i = (long)blockIdx.x * blockDim.x + threadIdx.x;
    if (i < n4) {
        float4 a = xv[i];
        float4 b = out[i];
        float4 r;
        r.x = fmaxf(a.x + b.x, 0.f);
        r.y = fmaxf(a.y + b.y, 0.f);
        r.z = fmaxf(a.z + b.z, 0.f);
        r.w = fmaxf(a.w + b.w, 0.f);
        out[i] = r;
    }
}

extern "C" void kernel_launch(void* const* d_in, const int* in_sizes, int n_in,
                              void* d_out, int out_size, void* d_ws, size_t ws_size,
                              hipStream_t stream)
{
    const float* x_v = (const float*)d_in[0];
    const float* x_c = (const float*)d_in[1];
    const float* W1v = (const float*)d_in[2];
    const float* b1v = (const float*)d_in[3];
    const float* W2v = (const float*)d_in[4];
    const float* b2v = (const float*)d_in[5];
    const float* W1c = (const float*)d_in[6];
    const float* b1c = (const float*)d_in[7];
    const float* W2c = (const float*)d_in[8];
    const float* b2c = (const float*)d_in[9];
    const int* src_vv = (const int*)d_in[10];
    const int* dst_vv = (const int*)d_in[11];
    const int* src_vc = (const int*)d_in[12];
    const int* dst_vc = (const int*)d_in[13];

    const int n   = in_sizes[0] / DD;   // 50000 vertices (multiple of 16)
    const int nc  = in_sizes[1] / DD;   // 256 colors
    const int Evv = in_sizes[10];
    const int Evc = in_sizes[12];

    float* Yv = (float*)d_ws;                 // n  x 64 f32
    float* Yc = Yv + (size_t)n * DD;          // nc x 64 f32
    float* agg = (float*)d_out;               // accumulate directly into output

    const long n4 = (long)n * DD / 4;

    zero_kernel<<<(int)((n4 + 255) / 256), 256, 0, stream>>>((float4*)agg, n4);

    // Per-node MLPs (WMMA f32): gather commutes with row-wise MLP.
    mlp16_wmma_kernel<<<n / 16, 32, 0, stream>>>(x_v, W1v, b1v, W2v, b2v, Yv);
    mlp16_wmma_kernel<<<nc / 16, 32, 0, stream>>>(x_c, W1c, b1c, W2c, b2c, Yc);

    // Edge scatter-sum of precomputed messages (L2-resident atomics).
    scatter_kernel<<<(Evv * 16 + 255) / 256, 256, 0, stream>>>(Yv, src_vv, dst_vv, agg, Evv);
    scatter_kernel<<<(Evc * 16 + 255) / 256, 256, 0, stream>>>(Yc, src_vc, dst_vc, agg, Evc);

    finalize_kernel<<<(int)((n4 + 255) / 256), 256, 0, stream>>>((const float4*)x_v,
                                                                 (float4*)agg, n4);
}